// QuantumQLSTM_65481071395816
// MI455X (gfx1250) — compile-verified
//
#include <hip/hip_runtime.h>

typedef float v2f __attribute__((ext_vector_type(2)));
typedef float v8f __attribute__((ext_vector_type(8)));

#define T_STEPS 512
#define BATCH   4096
#define DIN     32
#define NQ      4

// ---------------------------------------------------------------------------
// Phase 1: pre[t,b,n] = x[t,b,:] @ Wx[:,n] + b[n] + th[n],  n = gate*4 + wire
// One wave computes a 16(row) x 16(col) tile with 8x V_WMMA_F32_16X16X4_F32.
// ---------------------------------------------------------------------------
__global__ __launch_bounds__(256)
void qlstm_pre_gemm(const float* __restrict__ x,
                    const float* __restrict__ Wf, const float* __restrict__ Wi,
                    const float* __restrict__ Wu, const float* __restrict__ Wo,
                    const float* __restrict__ bf, const float* __restrict__ bi,
                    const float* __restrict__ bu, const float* __restrict__ bo,
                    const float* __restrict__ tf, const float* __restrict__ ti,
                    const float* __restrict__ tu, const float* __restrict__ to,
                    float* __restrict__ pre, int numTiles)
{
    const int lane   = threadIdx.x & 31;
    const int wave   = blockIdx.x * (blockDim.x >> 5) + (threadIdx.x >> 5);
    const int nWaves = gridDim.x * (blockDim.x >> 5);
    const int n  = lane & 15;   // output column (N) for B/C fragments
    const int hb = lane >> 4;   // half-wave selector
    const int g  = n >> 2;      // gate index 0..3 (f,i,u,o)
    const int w  = n & 3;       // wire index

    const float* Ws[4] = {Wf, Wi, Wu, Wo};
    const float* bs[4] = {bf, bi, bu, bo};
    const float* ts[4] = {tf, ti, tu, to};
    const float* Wg = Ws[g];

    // B fragments: chunk kc covers K = kc*4 .. kc*4+3 of the 32x16 Wx matrix.
    // Lanes 0-15 hold K = kc*4+{0,1}; lanes 16-31 hold K = kc*4+{2,3}.
    v2f bfrag[8];
#pragma unroll
    for (int kc = 0; kc < 8; ++kc) {
        int k0 = kc * 4 + 2 * hb;
        bfrag[kc].x = Wg[(k0 + 0) * NQ + w];
        bfrag[kc].y = Wg[(k0 + 1) * NQ + w];
    }
    const float bt = bs[g][w] + ts[g][w];   // bias + theta folded into C init

    for (int tile = wave; tile < numTiles; tile += nWaves) {
        const float* xrow = x + (size_t)(tile * 16 + n) * DIN + 2 * hb;
        v8f acc = {bt, bt, bt, bt, bt, bt, bt, bt};
#pragma unroll
        for (int kc = 0; kc < 8; ++kc) {
            v2f a = *(const v2f*)(xrow + kc * 4);
            acc = __builtin_amdgcn_wmma_f32_16x16x4_f32(
                false, a, false, bfrag[kc], (short)0, acc, false, false);
        }
        // C layout: VGPR r, lanes 0-15 -> row r, lanes 16-31 -> row 8+r, col n
        float* dst = pre + (size_t)tile * 256 + hb * 128 + n;
#pragma unroll
        for (int r = 0; r < 8; ++r)
            dst[r * 16] = acc[r];
    }
}

// ---------------------------------------------------------------------------
// Phase 2: per-batch-row recurrent scan. z-vector of each quantum gate is a
// product of cosines (closed form of the RX + CNOT-ring + <Z> circuit).
// ---------------------------------------------------------------------------
__device__ __forceinline__ float fast_sigmoid(float z) {
    return __builtin_amdgcn_rcpf(1.f + __expf(-z));
}
__device__ __forceinline__ float fast_tanh(float z) {
    float e = __expf(-2.f * z);
    return (1.f - e) * __builtin_amdgcn_rcpf(1.f + e);
}

__global__ __launch_bounds__(256)
void qlstm_scan(const float* __restrict__ pre,
                const float* __restrict__ Wf, const float* __restrict__ Wi,
                const float* __restrict__ Wu, const float* __restrict__ Wo,
                float* __restrict__ out)
{
    const int b = blockIdx.x * blockDim.x + threadIdx.x;
    if (b >= BATCH) return;

    // Hidden-part weights wh[j][n] = W_gate[(32+j)*4 + wire]; wave-uniform ->
    // compiler keeps these in SGPRs.
    const float* Ws[4] = {Wf, Wi, Wu, Wo};
    float wh[4][16];
#pragma unroll
    for (int g = 0; g < 4; ++g)
#pragma unroll
        for (int j = 0; j < 4; ++j)
#pragma unroll
            for (int w = 0; w < 4; ++w)
                wh[j][4 * g + w] = Ws[g][(DIN + j) * NQ + w];

    float hx[NQ] = {0.f, 0.f, 0.f, 0.f};
    float cx[NQ] = {0.f, 0.f, 0.f, 0.f};

    for (int t = 0; t < T_STEPS; ++t) {
        const float4* p = (const float4*)(pre + ((size_t)t * BATCH + b) * 16);
        float4 v0 = p[0], v1 = p[1], v2 = p[2], v3 = p[3];
        float a[16] = {v0.x, v0.y, v0.z, v0.w, v1.x, v1.y, v1.z, v1.w,
                       v2.x, v2.y, v2.z, v2.w, v3.x, v3.y, v3.z, v3.w};
        // add h @ Wh, then take cos of the full angle
#pragma unroll
        for (int n = 0; n < 16; ++n) {
            float v = a[n];
#pragma unroll
            for (int j = 0; j < NQ; ++j) v = fmaf(hx[j], wh[j][n], v);
            a[n] = __cosf(v);
        }
        // z expectations per gate from cosine products, then activations
        float gate[4][4];
#pragma unroll
        for (int g = 0; g < 4; ++g) {
            float C0 = a[4*g+0], C1 = a[4*g+1], C2 = a[4*g+2], C3 = a[4*g+3];
            float z1  = C0 * C1;
            float c23 = C2 * C3;
            float z[4] = {C1 * c23, z1, z1 * C2, z1 * c23};
#pragma unroll
            for (int w = 0; w < 4; ++w)
                gate[g][w] = (g == 2) ? fast_tanh(z[w]) : fast_sigmoid(z[w]);
        }
#pragma unroll
        for (int w = 0; w < 4; ++w) {
            cx[w] = fmaf(gate[0][w], cx[w], gate[1][w] * gate[2][w]);
            hx[w] = gate[3][w] * fast_tanh(cx[w]);
        }
        float4 ho = {hx[0], hx[1], hx[2], hx[3]};
        *(float4*)(out + ((size_t)t * BATCH + b) * NQ) = ho;
    }
    float4 h4 = {hx[0], hx[1], hx[2], hx[3]};
    float4 c4 = {cx[0], cx[1], cx[2], cx[3]};
    float* tail = out + (size_t)T_STEPS * BATCH * NQ;
    *(float4*)(tail + (size_t)b * NQ) = h4;
    *(float4*)(tail + (size_t)BATCH * NQ + (size_t)b * NQ) = c4;
}

// ---------------------------------------------------------------------------
extern "C" void kernel_launch(void* const* d_in, const int* in_sizes, int n_in,
                              void* d_out, int out_size, void* d_ws, size_t ws_size,
                              hipStream_t stream) {
    const float* x  = (const float*)d_in[0];
    const float* Wf = (const float*)d_in[1];
    const float* bf = (const float*)d_in[2];
    const float* tf = (const float*)d_in[3];
    const float* Wi = (const float*)d_in[4];
    const float* bi = (const float*)d_in[5];
    const float* ti = (const float*)d_in[6];
    const float* Wu = (const float*)d_in[7];
    const float* bu = (const float*)d_in[8];
    const float* tu = (const float*)d_in[9];
    const float* Wo = (const float*)d_in[10];
    const float* bo = (const float*)d_in[11];
    const float* to = (const float*)d_in[12];

    float* pre = (float*)d_ws;              // [T, B, 16] fp32 = 134 MB
    float* out = (float*)d_out;

    const int numTiles = (T_STEPS * BATCH) / 16;   // 131072 16-row tiles
    qlstm_pre_gemm<<<2048, 256, 0, stream>>>(x, Wf, Wi, Wu, Wo,
                                             bf, bi, bu, bo,
                                             tf, ti, tu, to, pre, numTiles);
    qlstm_scan<<<BATCH / 256, 256, 0, stream>>>(pre, Wf, Wi, Wu, Wo, out);
}